// LightGCN_40613210751549
// MI455X (gfx1250) — compile-verified
//
#include <hip/hip_runtime.h>

// ---------------------------------------------------------------------------
// LightGCN on MI455X (gfx1250).
// No dense GEMM exists in this workload (no weight matrices), so WMMA does not
// apply; the performance path is the memory system. Working set (~147 MB:
// node_f 38.4MB + agg 38.4MB + edges 32MB + res 38.4MB) fits in the 192 MB L2,
// so the 3-layer loop is L2-gather/atomic bound. We use 128-bit gathers,
// no-return global_atomic_add_f32 scatter, global_load_async_to_lds_b128 for
// the edge-index stream (ASYNCcnt path), and global_prefetch_b8 lookahead.
// ---------------------------------------------------------------------------

constexpr int USER_NUM  = 100000;
constexpr int ITEM_NUM  = 50000;
constexpr int N_NODES   = USER_NUM + ITEM_NUM;
constexpr int HIDE_DIM  = 64;
constexpr int N_EDGES   = 4000000;
constexpr int LAYER_NUM = 3;

constexpr int T4 = N_NODES * HIDE_DIM / 4;   // float4 count of a node-feature buffer
constexpr int U4 = USER_NUM * HIDE_DIM / 4;  // float4 count of user block

constexpr int EDGES_PER_BLOCK = 16;          // 256 threads / 16 lanes-per-edge

// No-return f32 atomic add -> global_atomic_add_f32 (result discarded, relaxed,
// agent scope: clang emits the no-rtn hardware atomic, tracked by STOREcnt).
__device__ __forceinline__ void atomic_add_f32(float* p, float v) {
  (void)__hip_atomic_fetch_add(p, v, __ATOMIC_RELAXED, __HIP_MEMORY_SCOPE_AGENT);
}

__global__ void zero_f4_kernel(float4* __restrict__ p, int n4) {
  int i = blockIdx.x * blockDim.x + threadIdx.x;
  if (i < n4) p[i] = make_float4(0.f, 0.f, 0.f, 0.f);
}

// out_deg[src[e]] += 1 ; in_deg[dst[e]] += 1
__global__ void deg_kernel(const int* __restrict__ src, const int* __restrict__ dst,
                           float* __restrict__ out_deg, float* __restrict__ in_deg) {
  int e = blockIdx.x * blockDim.x + threadIdx.x;
  if (e >= N_EDGES) return;
  atomic_add_f32(&out_deg[src[e]], 1.0f);
  atomic_add_f32(&in_deg[dst[e]],  1.0f);
}

// in place: deg -> max(deg,1)^-0.5
__global__ void norm_kernel(float* __restrict__ deg, int n) {
  int i = blockIdx.x * blockDim.x + threadIdx.x;
  if (i >= n) return;
  float v = deg[i];
  v = (v < 1.0f) ? 1.0f : v;
  deg[i] = 1.0f / sqrtf(v);
}

// node_f = concat(user_emb, item_emb); res(d_out) = same
__global__ void init_kernel(const float4* __restrict__ user4,
                            const float4* __restrict__ item4,
                            float4* __restrict__ node_f4,
                            float4* __restrict__ res4) {
  int i = blockIdx.x * blockDim.x + threadIdx.x;
  if (i >= T4) return;
  float4 v = (i < U4) ? user4[i] : item4[i - U4];
  node_f4[i] = v;
  res4[i]    = v;
}

// For each edge: agg[dst] += node_f[src] * out_norm[src].
// 16 lanes per edge, one float4 (16B) per lane -> full 64-float row.
// Edge indices for the block are DMA-staged into LDS via the gfx1250
// async-to-LDS path (ASYNCcnt), then broadcast-read from LDS.
__global__ __launch_bounds__(256) void scatter_kernel(
    const int* __restrict__ src, const int* __restrict__ dst,
    const float* __restrict__ node_f, const float* __restrict__ out_norm,
    float* __restrict__ agg) {
  __shared__ int edg[2 * EDGES_PER_BLOCK];       // [0..15]=src, [16..31]=dst
  const int e0 = blockIdx.x * EDGES_PER_BLOCK;   // N_EDGES % 16 == 0, no tail

  // Lanes 0..7 (wave 0) issue two async 128B-total copies: 4 lanes x b128 for
  // src indices, 4 lanes x b128 for dst indices.
  if (threadIdx.x < 8) {
    const int which = threadIdx.x >> 2;          // 0 = src, 1 = dst
    const int part  = threadIdx.x & 3;           // which 16B chunk
    const int* g = (which ? dst : src) + e0 + part * 4;
    auto* lp = (__attribute__((address_space(3))) int*)
                   &edg[which * EDGES_PER_BLOCK + part * 4];
    unsigned lds_off = (unsigned)(unsigned long long)lp;
    asm volatile("global_load_async_to_lds_b128 %0, %1, off"
                 :: "v"(lds_off), "v"(g) : "memory");
  }
  asm volatile("s_wait_asynccnt 0x0" ::: "memory");  // wave-local DMA complete
  __syncthreads();                                   // publish LDS to all waves

  const int eL   = threadIdx.x >> 4;   // 0..15: edge within block
  const int lane = threadIdx.x & 15;   // 0..15: float4 within the 64-f row
  const int e    = e0 + eL;

  // Prefetch the edge-index stream ahead of use (global_prefetch_b8,
  // speculative). Indices are the only perfectly-sequential traffic.
  int ep = e + 4096;
  if (ep >= N_EDGES) ep = N_EDGES - 1;
  __builtin_prefetch(&src[ep], 0, 0);
  __builtin_prefetch(&dst[ep], 0, 0);

  const int   s  = edg[eL];
  const int   d  = edg[EDGES_PER_BLOCK + eL];
  const float on = out_norm[s];

  const float4 v = *(const float4*)(node_f + (size_t)s * HIDE_DIM + lane * 4);
  float* ap = agg + (size_t)d * HIDE_DIM + lane * 4;
  atomic_add_f32(ap + 0, v.x * on);
  atomic_add_f32(ap + 1, v.y * on);
  atomic_add_f32(ap + 2, v.z * on);
  atomic_add_f32(ap + 3, v.w * on);
}

// emb = agg * in_norm[node]; node_f = emb; res += emb * scale.
// Also re-zeroes agg in the same pass (saves a 38 MB zero kernel per layer).
__global__ void update_kernel(float4* __restrict__ agg4,
                              const float* __restrict__ in_norm,
                              float4* __restrict__ node_f4,
                              float4* __restrict__ res4,
                              float scale) {
  int i = blockIdx.x * blockDim.x + threadIdx.x;
  if (i >= T4) return;
  int   node = i >> 4;            // 16 float4 per node row
  float inn  = in_norm[node];
  float4 a = agg4[i];
  agg4[i] = make_float4(0.f, 0.f, 0.f, 0.f);   // ready for next layer's scatter
  float4 emb = make_float4(a.x * inn, a.y * inn, a.z * inn, a.w * inn);
  node_f4[i] = emb;
  float4 r = res4[i];
  r.x += emb.x * scale;
  r.y += emb.y * scale;
  r.z += emb.z * scale;
  r.w += emb.w * scale;
  res4[i] = r;
}

extern "C" void kernel_launch(void* const* d_in, const int* in_sizes, int n_in,
                              void* d_out, int out_size, void* d_ws, size_t ws_size,
                              hipStream_t stream) {
  const float* user_emb = (const float*)d_in[0];   // [USER_NUM, 64]
  const float* item_emb = (const float*)d_in[1];   // [ITEM_NUM, 64]
  const int*   src      = (const int*)d_in[2];     // [N_EDGES]
  const int*   dst      = (const int*)d_in[3];     // [N_EDGES]
  float*       res      = (float*)d_out;           // [N_NODES, 64] concat(user,item)

  // Workspace layout (floats):
  //   [0, N)              out_norm / out_deg
  //   [N, 2N)             in_norm  / in_deg
  //   [2N, 2N + N*64)     node_f
  //   [.., +N*64)         agg
  float* ws       = (float*)d_ws;
  float* out_norm = ws;
  float* in_norm  = ws + N_NODES;
  float* node_f   = ws + 2 * (size_t)N_NODES;
  float* agg      = node_f + (size_t)N_NODES * HIDE_DIM;

  const int TPB = 256;
  auto blocks = [](long long n, int tpb) { return (int)((n + tpb - 1) / tpb); };

  // 1) Degrees -> symmetric norms. (2*N_NODES floats = 75000 float4, exact.)
  const int NORM4 = 2 * N_NODES / 4;
  zero_f4_kernel<<<blocks(NORM4, TPB), TPB, 0, stream>>>((float4*)out_norm, NORM4);
  deg_kernel<<<blocks(N_EDGES, TPB), TPB, 0, stream>>>(src, dst, out_norm, in_norm);
  norm_kernel<<<blocks(2 * N_NODES, TPB), TPB, 0, stream>>>(out_norm, 2 * N_NODES);

  // 2) node_f = concat(user, item); res = concat(user, item); agg = 0 (once —
  //    update_kernel re-zeroes it each layer as it consumes it).
  init_kernel<<<blocks(T4, TPB), TPB, 0, stream>>>(
      (const float4*)user_emb, (const float4*)item_emb,
      (float4*)node_f, (float4*)res);
  zero_f4_kernel<<<blocks(T4, TPB), TPB, 0, stream>>>((float4*)agg, T4);

  // 3) Three propagation layers, all L2-resident on MI455X (192 MB L2).
  const int scatter_blocks = N_EDGES / EDGES_PER_BLOCK;  // 250000, exact
  for (int layer = 0; layer < LAYER_NUM; ++layer) {
    scatter_kernel<<<scatter_blocks, TPB, 0, stream>>>(
        src, dst, node_f, out_norm, agg);

    float scale = 1.0f / (float)(layer + 2);
    update_kernel<<<blocks(T4, TPB), TPB, 0, stream>>>(
        (float4*)agg, in_norm, (float4*)node_f, (float4*)res, scale);
  }
}